// SpectralConv_15650860827413
// MI455X (gfx1250) — compile-verified
//
#include <hip/hip_runtime.h>

typedef float v2f __attribute__((ext_vector_type(2)));
typedef float v8f __attribute__((ext_vector_type(8)));

#define B_    2
#define C_    256
#define H_    256
#define W_    512
#define MW_   257          // W/2+1 rfft bins
#define NCOL_ 514          // B * MW_
#define PI_   3.14159265358979323846f

// ---------------------------------------------------------------------------
// Stockham radix-2 autosort FFT, N = 256, ping-pong a<->b, result lands in a.
// COLS = interleave stride (multi-column batched FFT in LDS).
// sign = -1 forward, +1 inverse (unnormalized).
// ---------------------------------------------------------------------------
template <int COLS>
__device__ __forceinline__ void stockham256(float2* a, float2* b, int col,
                                            int bfStart, int bfCount, float sign) {
  float2* src = a;
  float2* dst = b;
#pragma unroll
  for (int s = 0; s < 8; ++s) {
    int mst = 1 << s;
    __syncthreads();
    for (int q = 0; q < bfCount; ++q) {
      int bf = bfStart + q;           // butterfly id in [0,128)
      int j  = bf >> s;
      int k  = bf & (mst - 1);
      float2 c0 = src[(k + j * mst) * COLS + col];
      float2 c1 = src[(k + j * mst + 128) * COLS + col];
      float th = (float)(j * mst) * (PI_ / 128.0f);
      float sn, cs;
      __sincosf(th, &sn, &cs);
      sn *= sign;                      // w = cs + i*sn
      float2 d0 = make_float2(c0.x + c1.x, c0.y + c1.y);
      float tx = c0.x - c1.x, ty = c0.y - c1.y;
      float2 d1 = make_float2(tx * cs - ty * sn, tx * sn + ty * cs);
      dst[(k + 2 * j * mst) * COLS + col]       = d0;
      dst[(k + 2 * j * mst + mst) * COLS + col] = d1;
    }
    float2* t = src; src = dst; dst = t;
  }
  __syncthreads();
}

// ---------------------------------------------------------------------------
// Stage 1: rfft along W (512 real -> 257 complex) via packed 256-pt CFFT.
// One block (128 thr) per row (b,c,h). Output planes Fr/Fi [b][c][h][m].
// ---------------------------------------------------------------------------
__global__ __launch_bounds__(128) void fftw_kernel(const float* __restrict__ x,
                                                   float* __restrict__ Fr,
                                                   float* __restrict__ Fi) {
  __shared__ float2 sa[256], sb[256];
  int row = blockIdx.x;
  int t   = threadIdx.x;
  const float2* xr = (const float2*)(x + (size_t)row * W_);  // z[n]=x[2n]+i x[2n+1]
  sa[t]       = xr[t];
  sa[t + 128] = xr[t + 128];
  stockham256<1>(sa, sb, 0, t, 1, -1.0f);
  size_t ob = (size_t)row * MW_;
#pragma unroll
  for (int u = 0; u < 2; ++u) {
    int k = t + 128 * u;
    float2 Zk = sa[k];
    float2 Zm = sa[(256 - k) & 255];
    float Er  = 0.5f * (Zk.x + Zm.x), Ei = 0.5f * (Zk.y - Zm.y);
    float Or_ = 0.5f * (Zk.x - Zm.x), Oi = 0.5f * (Zk.y + Zm.y);
    float th = (float)k * (PI_ / 256.0f);   // 2*pi*k/512
    float sn, cs;
    __sincosf(th, &sn, &cs);
    Fr[ob + k] = Er - sn * Or_ + cs * Oi;   // X = E - i e^{-i th} O
    Fi[ob + k] = Ei - cs * Or_ - sn * Oi;
  }
  if (t == 0) {                              // Nyquist bin m = 256
    float2 Z0 = sa[0];
    Fr[ob + 256] = Z0.x - Z0.y;
    Fi[ob + 256] = 0.0f;
  }
}

// ---------------------------------------------------------------------------
// Stage 2 / 4: batched 256-pt complex FFT along H for 16 m-columns per block.
// Block (16,16): x = column lane, y = butterfly worker. LDS layout [idx][col]
// -> bank-conflict-free, global loads/stores coalesced along m.
// ---------------------------------------------------------------------------
__global__ __launch_bounds__(256) void ffth_kernel(const float* __restrict__ inR,
                                                   const float* __restrict__ inI,
                                                   float* __restrict__ outR,
                                                   float* __restrict__ outI,
                                                   float sign, float scale) {
  __shared__ float2 sa[256 * 16], sb[256 * 16];
  int bc = blockIdx.y;            // flattened (b, channel)
  int m0 = blockIdx.x * 16;
  int xl = threadIdx.x;           // column lane 0..15
  int yw = threadIdx.y;           // worker 0..15
  int m  = m0 + xl;
  bool valid  = (m < MW_);
  size_t base = (size_t)bc * H_ * MW_ + m;
#pragma unroll
  for (int it = 0; it < 16; ++it) {
    int h = yw + 16 * it;
    float2 v = make_float2(0.0f, 0.0f);
    if (valid) {
      size_t g = base + (size_t)h * MW_;
      v.x = inR[g];
      v.y = inI[g];
    }
    sa[h * 16 + xl] = v;
  }
  stockham256<16>(sa, sb, xl, yw * 8, 8, sign);
  if (valid) {
#pragma unroll
    for (int it = 0; it < 16; ++it) {
      int l = yw + 16 * it;
      float2 v = sa[l * 16 + xl];
      size_t g = base + (size_t)l * MW_;
      outR[g] = v.x * scale;
      outI[g] = v.y * scale;
    }
  }
}

// ---------------------------------------------------------------------------
// Weight transpose: w[i][o][l] -> wT[l][o][i] (real & imag planes), LDS-tiled
// so both global read and write are coalesced.
// ---------------------------------------------------------------------------
__global__ __launch_bounds__(256) void wtrans_kernel(const float* __restrict__ wr,
                                                     const float* __restrict__ wi,
                                                     float* __restrict__ tr,
                                                     float* __restrict__ ti) {
  __shared__ float sr[32][33], si[32][33];
  int i0 = blockIdx.x * 32;
  int l0 = blockIdx.y * 32;
  int o  = blockIdx.z;
  int xl = threadIdx.x;   // 0..31
  int yw = threadIdx.y;   // 0..7
#pragma unroll
  for (int it = 0; it < 4; ++it) {
    int i = yw + 8 * it;
    size_t g = ((size_t)(i0 + i) * C_ + o) * H_ + l0 + xl;   // lanes along l
    sr[i][xl] = wr[g];
    si[i][xl] = wi[g];
  }
  __syncthreads();
#pragma unroll
  for (int it = 0; it < 4; ++it) {
    int l = yw + 8 * it;
    size_t g = ((size_t)(l0 + l) * C_ + o) * C_ + i0 + xl;   // lanes along i
    tr[g] = sr[xl][l];
    ti[g] = si[xl][l];
  }
}

// ---------------------------------------------------------------------------
// Stage 3: per-l complex GEMM  Y[o,n] = sum_i wT[l][o][i] * X[i,n]
// M = Cout = 256 (2 blocks of 128), N = 514 cols (17 tiles of 32), K = 256.
// 8 waves: wave w owns rows 16w..16w+15 and both 16-wide N subtiles.
// v_wmma_f32_16x16x4_f32, 4 accumulators per subtile (RR, II, RI, IR):
// f32 WMMA has no A-negate, so real = RR-II / imag = RI+IR combined at end.
// LDS K-stride padded to 34 floats: 8B-aligned b64 fragment loads, no bank
// conflicts (17 odd => n*34 mod 64 distinct across 32 lanes).
// ---------------------------------------------------------------------------
#define KT  32
#define LDK 34

__global__ __launch_bounds__(256) void gemm_kernel(const float* __restrict__ wTr,
                                                   const float* __restrict__ wTi,
                                                   const float* __restrict__ Xr,
                                                   const float* __restrict__ Xi,
                                                   float* __restrict__ Yr,
                                                   float* __restrict__ Yi) {
  __shared__ float Asr[128 * LDK], Asi[128 * LDK];
  __shared__ float Bsr[32 * LDK],  Bsi[32 * LDK];
  int nB    = blockIdx.x * 32;
  int oB    = blockIdx.y * 128;
  int lmode = blockIdx.z;
  int tid  = threadIdx.x;
  int lane = tid & 31, wv = tid >> 5;
  int lrow = lane & 15, lhi = lane >> 4;

  v8f aRR0 = {}, aII0 = {}, aRI0 = {}, aIR0 = {};
  v8f aRR1 = {}, aII1 = {}, aRI1 = {}, aIR1 = {};

  for (int k0 = 0; k0 < C_; k0 += KT) {
    __syncthreads();
    {  // stage A tile: 128(o) x 32(k), lanes along k (coalesced)
      int kk = tid & 31;
      int ob = tid >> 5;
      size_t wbase = ((size_t)lmode * C_ + oB) * C_ + (size_t)(k0 + kk);
#pragma unroll
      for (int it = 0; it < 16; ++it) {
        int o = ob + 8 * it;
        Asr[o * LDK + kk] = wTr[wbase + (size_t)o * C_];
        Asi[o * LDK + kk] = wTi[wbase + (size_t)o * C_];
      }
    }
    {  // stage B tile: 32(n) x 32(k), lanes along n (coalesced along m)
      int nl = tid & 31;
      int kb = tid >> 5;
      int n  = nB + nl;
      bool ok = (n < NCOL_);
      int bb = n / MW_, mm = n % MW_;
#pragma unroll
      for (int it = 0; it < 4; ++it) {
        int kk = kb + 8 * it;
        float vr = 0.0f, vi = 0.0f;
        if (ok) {
          size_t g = (((size_t)bb * C_ + (size_t)(k0 + kk)) * H_ + lmode) * MW_ + mm;
          vr = Xr[g];
          vi = Xi[g];
        }
        Bsr[nl * LDK + kk] = vr;
        Bsi[nl * LDK + kk] = vi;
      }
    }
    __syncthreads();

    // A frag (16x4): lanes 0-15 M rows, K pair {0,1} lo / {2,3} hi lanes.
    int aoff  = (wv * 16 + lrow) * LDK + 2 * lhi;
    int b0off = lrow * LDK + 2 * lhi;
    int b1off = (16 + lrow) * LDK + 2 * lhi;
#pragma unroll
    for (int kk = 0; kk < KT; kk += 4) {
      v2f ar  = *(const v2f*)&Asr[aoff + kk];
      v2f ai  = *(const v2f*)&Asi[aoff + kk];
      v2f br0 = *(const v2f*)&Bsr[b0off + kk];
      v2f bi0 = *(const v2f*)&Bsi[b0off + kk];
      v2f br1 = *(const v2f*)&Bsr[b1off + kk];
      v2f bi1 = *(const v2f*)&Bsi[b1off + kk];
      aRR0 = __builtin_amdgcn_wmma_f32_16x16x4_f32(false, ar, false, br0, (short)0, aRR0, false, false);
      aII0 = __builtin_amdgcn_wmma_f32_16x16x4_f32(false, ai, false, bi0, (short)0, aII0, false, false);
      aRI0 = __builtin_amdgcn_wmma_f32_16x16x4_f32(false, ar, false, bi0, (short)0, aRI0, false, false);
      aIR0 = __builtin_amdgcn_wmma_f32_16x16x4_f32(false, ai, false, br0, (short)0, aIR0, false, false);
      aRR1 = __builtin_amdgcn_wmma_f32_16x16x4_f32(false, ar, false, br1, (short)0, aRR1, false, false);
      aII1 = __builtin_amdgcn_wmma_f32_16x16x4_f32(false, ai, false, bi1, (short)0, aII1, false, false);
      aRI1 = __builtin_amdgcn_wmma_f32_16x16x4_f32(false, ar, false, bi1, (short)0, aRI1, false, false);
      aIR1 = __builtin_amdgcn_wmma_f32_16x16x4_f32(false, ai, false, br1, (short)0, aIR1, false, false);
    }
  }
  v8f yr0 = aRR0 - aII0, yi0 = aRI0 + aIR0;
  v8f yr1 = aRR1 - aII1, yi1 = aRI1 + aIR1;

  // C/D layout: VGPR r -> M = r + 8*lhi, N = lrow (per 16-wide subtile).
  int orow = oB + wv * 16 + 8 * lhi;
#pragma unroll
  for (int sub = 0; sub < 2; ++sub) {
    int n = nB + sub * 16 + lrow;
    if (n < NCOL_) {
      int bb = n / MW_, mm = n % MW_;
      v8f yr = sub ? yr1 : yr0;
      v8f yi = sub ? yi1 : yi0;
#pragma unroll
      for (int r = 0; r < 8; ++r) {
        size_t g = (((size_t)bb * C_ + (size_t)(orow + r)) * H_ + lmode) * MW_ + mm;
        Yr[g] = yr[r];
        Yi[g] = yi[r];
      }
    }
  }
}

// ---------------------------------------------------------------------------
// Stage 5: inverse rfft along W (257 complex -> 512 real) + bias.
// Hermitian pack into 256-pt inverse CFFT; normalization 1/512 for this axis
// (the /2 in E,O recovery plus 1/256 on the IFFT).
// ---------------------------------------------------------------------------
__global__ __launch_bounds__(128) void ifftw_kernel(const float* __restrict__ Gr,
                                                    const float* __restrict__ Gi,
                                                    const float* __restrict__ bias,
                                                    float* __restrict__ out) {
  __shared__ float2 sa[256], sb[256];
  int row = blockIdx.x;            // (b, o, h) flattened
  int t   = threadIdx.x;
  int o   = (row >> 8) & 255;
  size_t ib = (size_t)row * MW_;
#pragma unroll
  for (int u = 0; u < 2; ++u) {
    int k = t + 128 * u;
    float Xr0 = Gr[ib + k],       Xi0 = Gi[ib + k];
    float Xr1 = Gr[ib + 256 - k], Xi1 = Gi[ib + 256 - k];
    float Er = 0.5f * (Xr0 + Xr1), Ei = 0.5f * (Xi0 - Xi1);
    float Dr = 0.5f * (Xr0 - Xr1), Di = 0.5f * (Xi0 + Xi1);
    float th = (float)k * (PI_ / 256.0f);
    float sn, cs;
    __sincosf(th, &sn, &cs);
    float Or_ = cs * Dr - sn * Di;           // O = e^{+i th} D
    float Oi  = sn * Dr + cs * Di;
    sa[k] = make_float2(Er - Oi, Ei + Or_);  // Z = E + i O
  }
  stockham256<1>(sa, sb, 0, t, 1, 1.0f);
  float bv = bias[o];
  float2* op = (float2*)(out + (size_t)row * W_);
  const float inv = 1.0f / 256.0f;
#pragma unroll
  for (int u = 0; u < 2; ++u) {
    int k = t + 128 * u;
    float2 z = sa[k];
    op[k] = make_float2(z.x * inv + bv, z.y * inv + bv);  // y[2k], y[2k+1]
  }
}

// ---------------------------------------------------------------------------
// Residual copy (second output): float4 streaming copy of x.
// ---------------------------------------------------------------------------
__global__ __launch_bounds__(256) void residual_kernel(const float* __restrict__ x,
                                                       float* __restrict__ out) {
  size_t i = (size_t)blockIdx.x * 256 + threadIdx.x;
  ((float4*)out)[i] = ((const float4*)x)[i];
}

// ---------------------------------------------------------------------------
extern "C" void kernel_launch(void* const* d_in, const int* in_sizes, int n_in,
                              void* d_out, int out_size, void* d_ws, size_t ws_size,
                              hipStream_t stream) {
  const float* x    = (const float*)d_in[0];
  const float* wr   = (const float*)d_in[1];
  const float* wi   = (const float*)d_in[2];
  const float* bias = (const float*)d_in[3];
  float* y   = (float*)d_out;
  float* res = y + (size_t)B_ * C_ * H_ * W_;

  // Workspace layout (floats). Needs 2*NW + 4*NF floats (~673 MB).
  float* ws = (float*)d_ws;
  const size_t NW = (size_t)C_ * C_ * H_;          // transposed weight plane
  const size_t NF = (size_t)B_ * C_ * H_ * MW_;    // spectral plane
  float* wTr = ws;
  float* wTi = ws + NW;
  float* Fr  = ws + 2 * NW;       // fwd-W spectrum; reused for Y after GEMM
  float* Fi  = Fr + NF;
  float* Xr  = Fi + NF;           // full 2D spectrum; reused for G after iFFT-H
  float* Xi  = Xr + NF;

  // residual = x  (independent of the spectral pipeline)
  residual_kernel<<<(B_ * C_ * H_ * W_) / (256 * 4), 256, 0, stream>>>(x, res);
  // weight transpose -> [l][o][i]
  wtrans_kernel<<<dim3(8, 8, 256), dim3(32, 8), 0, stream>>>(wr, wi, wTr, wTi);
  // forward rfft along W
  fftw_kernel<<<B_ * C_ * H_, 128, 0, stream>>>(x, Fr, Fi);
  // forward cfft along H
  ffth_kernel<<<dim3(17, B_ * C_), dim3(16, 16), 0, stream>>>(Fr, Fi, Xr, Xi, -1.0f, 1.0f);
  // per-l complex channel mix (WMMA f32)
  gemm_kernel<<<dim3(17, 2, 256), 256, 0, stream>>>(wTr, wTi, Xr, Xi, Fr, Fi);
  // inverse cfft along H (scale 1/256)
  ffth_kernel<<<dim3(17, B_ * C_), dim3(16, 16), 0, stream>>>(Fr, Fi, Xr, Xi, 1.0f, 1.0f / 256.0f);
  // inverse rfft along W + bias
  ifftw_kernel<<<B_ * C_ * H_, 128, 0, stream>>>(Xr, Xi, bias, y);
}